// SpikeEncoder_36000415875202
// MI455X (gfx1250) — compile-verified
//
#include <hip/hip_runtime.h>

// Temporal spike encoder for MI455X (gfx1250).
// Pure store-bandwidth problem: 4 MB in -> 84 MB out (write-once stream).
// Strategy: float4 granularity per thread, 20 fully-unrolled nontemporal
// 128-bit stores (th:TH_STORE_NT) so the 84 MB bypasses L2 residency and
// saturates the 23.3 TB/s HBM write path. Wave32 lanes cover 128 consecutive
// neurons -> each of the 20 stores is a coalesced 512B burst per wave.

typedef float v4f __attribute__((ext_vector_type(4)));

#define N_TIMESTEPS 20
#define N_NEURONS   1024

__global__ __launch_bounds__(256) void
SpikeEncoder_36000415875202_kernel(const float* __restrict__ features,
                                   const float* __restrict__ latency_scale,
                                   float* __restrict__ out,
                                   int total_vec /* B*S*N/4 */) {
  const int idx = blockIdx.x * blockDim.x + threadIdx.x;
  if (idx >= total_vec) return;

  const int nvec = N_NEURONS / 4;          // 256 float4 per (b,s) row
  const int bs   = idx / nvec;             // flattened (b,s)
  const int n    = (idx - bs * nvec) * 4;  // neuron index (multiple of 4)

  // 128-bit loads of features and per-neuron scale (read-once, streaming).
  const v4f f  = __builtin_nontemporal_load(
      reinterpret_cast<const v4f*>(features + (size_t)bs * N_NEURONS + n));
  const v4f sc = *reinterpret_cast<const v4f*>(latency_scale + n);

  // lat = trunc((1 - clamp(f,0,1)) * T * scale); trunc matches astype(int32)
  // for the non-negative values produced here. lat==T (f<=0) emits no spike.
  const int l0 = (int)((1.0f - fminf(fmaxf(f.x, 0.0f), 1.0f)) * (float)N_TIMESTEPS * sc.x);
  const int l1 = (int)((1.0f - fminf(fmaxf(f.y, 0.0f), 1.0f)) * (float)N_TIMESTEPS * sc.y);
  const int l2 = (int)((1.0f - fminf(fmaxf(f.z, 0.0f), 1.0f)) * (float)N_TIMESTEPS * sc.z);
  const int l3 = (int)((1.0f - fminf(fmaxf(f.w, 0.0f), 1.0f)) * (float)N_TIMESTEPS * sc.w);

  // Output layout: [(b,s), t, n] -> stride N between timesteps.
  float* obase = out + ((size_t)bs * N_TIMESTEPS) * N_NEURONS + n;

#pragma unroll
  for (int t = 0; t < N_TIMESTEPS; ++t) {
    v4f v;
    v.x = (l0 == t) ? 1.0f : 0.0f;
    v.y = (l1 == t) ? 1.0f : 0.0f;
    v.z = (l2 == t) ? 1.0f : 0.0f;
    v.w = (l3 == t) ? 1.0f : 0.0f;
    // Non-temporal: output is write-once, never re-read by this kernel.
    __builtin_nontemporal_store(
        v, reinterpret_cast<v4f*>(obase + (size_t)t * N_NEURONS));
  }
}

extern "C" void kernel_launch(void* const* d_in, const int* in_sizes, int n_in,
                              void* d_out, int out_size, void* d_ws, size_t ws_size,
                              hipStream_t stream) {
  const float* features      = (const float*)d_in[0];  // [B, S, N] f32
  const float* latency_scale = (const float*)d_in[1];  // [N] f32
  float*       out           = (float*)d_out;          // [B, S, T, N] f32

  const int total_elems = in_sizes[0];      // B*S*N = 1,048,576
  const int total_vec   = total_elems / 4;  // 262,144 float4 threads

  const int block = 256;                    // 8 wave32 waves per block
  const int grid  = (total_vec + block - 1) / block;

  SpikeEncoder_36000415875202_kernel<<<grid, block, 0, stream>>>(
      features, latency_scale, out, total_vec);
}